// Topological_Loss_14705968022250
// MI455X (gfx1250) — compile-verified
//
#include <hip/hip_runtime.h>
#include <math.h>

#define Hh   128
#define Ww   128
#define NPIX 16384
#define BATCH 4
#define NIMG  8          // 0..3 = mask images, 4..7 = pred images
#define KPTS  256

typedef __attribute__((ext_vector_type(2))) float v2f;
typedef __attribute__((ext_vector_type(8))) float v8f;

// -------- workspace layout (bytes) --------
#define WS_MNMX   0                              // 4 samples * {mn,mx} floats
#define WS_MAXLEV 256                            // 8 ints
#define WS_LEV    512                            // 8*16384 u8  (birth level)
#define WS_DLEV   (WS_LEV  + NIMG*NPIX)          // 8*16384 u8  (death level)
#define WS_PERS   (WS_DLEV + NIMG*NPIX)          // 8*16384 u8  (persistence level)
#define WS_RANK   (WS_PERS + NIMG*NPIX)          // 8*16384 u16
#define WS_ORDER  (WS_RANK + NIMG*NPIX*2)        // 8*16384 u16
#define WS_BK     (WS_ORDER + NIMG*NPIX*2)       // 8*256 f32
#define WS_DK     (WS_BK + NIMG*KPTS*4)          // 8*256 f32
#define WS_PK     (WS_DK + NIMG*KPTS*4)          // 8*256 f32

// ---------------- kernel 1: per-sample min/max of sigmoid(pred) ----------------
__global__ void k1_minmax(const float* __restrict__ x, float* __restrict__ mnmx) {
  __shared__ float smin[256], smax[256];
  int b = blockIdx.x, tid = threadIdx.x;
  const float* p = x + b * NPIX;
  float mn = 1e30f, mx = -1e30f;
  for (int i = tid; i < NPIX; i += 256) { float v = p[i]; mn = fminf(mn, v); mx = fmaxf(mx, v); }
  smin[tid] = mn; smax[tid] = mx; __syncthreads();
  for (int s = 128; s > 0; s >>= 1) {
    if (tid < s) { smin[tid] = fminf(smin[tid], smin[tid + s]); smax[tid] = fmaxf(smax[tid], smax[tid + s]); }
    __syncthreads();
  }
  if (tid == 0) {
    // sigmoid is monotonic: min/max of sigmoid == sigmoid(min/max)
    mnmx[b * 2 + 0] = 1.f / (1.f + expf(-smin[0]));
    mnmx[b * 2 + 1] = 1.f / (1.f + expf(-smax[0]));
  }
}

// -------- kernel 2: quantize to 11 levels + stable counting-sort rank/order --------
__global__ void k2_quant_rank(const float* __restrict__ mo, const float* __restrict__ lb,
                              const float* __restrict__ mnmx,
                              unsigned char* __restrict__ levg,
                              unsigned short* __restrict__ rankg,
                              unsigned short* __restrict__ ordg,
                              int* __restrict__ maxlevg) {
  __shared__ unsigned int cnt[11][256];
  __shared__ unsigned int base[11], total[11];
  int img = blockIdx.x, tid = threadIdx.x;
  unsigned char*  lev = levg  + img * NPIX;
  unsigned short* rnk = rankg + img * NPIX;
  unsigned short* ord = ordg  + img * NPIX;
  int s = img & 3;
  bool isPred = img >= 4;
  float mn = 0.f, inv = 10.f;
  if (isPred) { mn = mnmx[s * 2]; float mx = mnmx[s * 2 + 1]; inv = 10.f / (mx - mn); }

  int c[11];
  #pragma unroll
  for (int L = 0; L < 11; ++L) c[L] = 0;
  for (int j = 0; j < 64; ++j) {
    int i = tid * 64 + j;
    float v;
    if (isPred) { float xx = mo[s * NPIX + i]; float sg = 1.f / (1.f + expf(-xx)); v = (sg - mn) * inv; }
    else        { v = lb[s * NPIX + i] * 10.f; }
    int L = (int)rintf(v);            // round-half-even == jnp.round
    L = L < 0 ? 0 : (L > 10 ? 10 : L);
    lev[i] = (unsigned char)L;
    c[L]++;
  }
  #pragma unroll
  for (int L = 0; L < 11; ++L) cnt[L][tid] = (unsigned)c[L];
  __syncthreads();
  if (tid < 11) {                       // exclusive scan per level across thread chunks
    unsigned run = 0;
    for (int t = 0; t < 256; ++t) { unsigned tmp = cnt[tid][t]; cnt[tid][t] = run; run += tmp; }
    total[tid] = run;
  }
  __syncthreads();
  if (tid == 0) {
    unsigned acc = 0; int mxl = 0;
    for (int L = 0; L < 11; ++L) { base[L] = acc; acc += total[L]; if (total[L] > 0) mxl = L; }
    maxlevg[img] = mxl;
  }
  __syncthreads();
  int c2[11];
  #pragma unroll
  for (int L = 0; L < 11; ++L) c2[L] = 0;
  for (int j = 0; j < 64; ++j) {        // rank = argsort(argsort(v)), stable by index
    int i = tid * 64 + j;
    int L = lev[i];
    unsigned r = base[L] + cnt[L][tid] + (unsigned)c2[L]; c2[L]++;
    rnk[i] = (unsigned short)r;
    ord[r] = (unsigned short)i;
  }
}

// -------- kernel 3: LDS-resident Kruskal union-find (elder rule), 1 image / WG --------
__global__ void k3_uf(const unsigned char* __restrict__ levg,
                      const unsigned short* __restrict__ rankg,
                      const unsigned short* __restrict__ ordg,
                      const int* __restrict__ maxlevg,
                      unsigned char* __restrict__ dlevg,
                      unsigned char* __restrict__ persg) {
  extern __shared__ unsigned short sm[];          // 4 * 32KB = 128KB dynamic LDS
  unsigned short* par = sm;
  unsigned short* rnk = sm + NPIX;
  unsigned short* ord = sm + 2 * NPIX;
  unsigned short* dth = sm + 3 * NPIX;
  int img = blockIdx.x, tid = threadIdx.x;
  const unsigned short* rg = rankg + img * NPIX;
  const unsigned short* og = ordg  + img * NPIX;
  for (int i = tid; i < NPIX; i += blockDim.x) {
    par[i] = (unsigned short)i;
    dth[i] = 0xFFFFu;
    rnk[i] = rg[i];
    ord[i] = og[i];
  }
  __syncthreads();
  if (tid == 0) {
    for (int t = 0; t < NPIX; ++t) {
      int p = ord[t];
      int row = p >> 7, col = p & 127;
      #pragma unroll
      for (int k = 0; k < 4; ++k) {
        int q;
        switch (k) {
          case 0:  q = (row > 0)      ? p - Ww : -1; break;
          case 1:  q = (row < Hh - 1) ? p + Ww : -1; break;
          case 2:  q = (col > 0)      ? p - 1  : -1; break;
          default: q = (col < Ww - 1) ? p + 1  : -1; break;
        }
        if (q < 0) continue;
        if ((int)rnk[q] >= t) continue;             // neighbor not yet entered
        int rp = p; while (par[rp] != rp) { unsigned short g = par[par[rp]]; par[rp] = g; rp = g; }
        int rq = q; while (par[rq] != rq) { unsigned short g = par[par[rq]]; par[rq] = g; rq = g; }
        if (rp == rq) continue;
        int older, young;
        if (rnk[rp] < rnk[rq]) { older = rp; young = rq; } else { older = rq; young = rp; }
        par[young] = (unsigned short)older;
        dth[young] = (unsigned short)p;             // death at current pixel
      }
    }
  }
  __syncthreads();
  const unsigned char* lev = levg + img * NPIX;
  unsigned char* dlev = dlevg + img * NPIX;
  unsigned char* prs  = persg + img * NPIX;
  int mxl = maxlevg[img];                           // essential class dies at max(v)
  for (int i = tid; i < NPIX; i += blockDim.x) {
    int b = lev[i];
    int d = dth[i];
    int dl = (d == 0xFFFF) ? mxl : (int)lev[d];
    dlev[i] = (unsigned char)dl;
    prs[i]  = (unsigned char)(dl > b ? dl - b : 0);  // valid <=> death > birth
  }
}

// -------- kernel 4: deterministic top-K by persistence (counting selection) --------
__global__ void k4_topk(const unsigned char* __restrict__ levg,
                        const unsigned char* __restrict__ dlevg,
                        const unsigned char* __restrict__ persg,
                        float* __restrict__ bk, float* __restrict__ dk, float* __restrict__ pk) {
  __shared__ unsigned int cnt[10][256];   // persistence levels 1..10 -> [L-1]
  __shared__ unsigned int above[11], total[10];
  int img = blockIdx.x, tid = threadIdx.x;
  const unsigned char* lev = levg  + img * NPIX;
  const unsigned char* dlv = dlevg + img * NPIX;
  const unsigned char* prs = persg + img * NPIX;
  float* b  = bk + img * KPTS;
  float* d  = dk + img * KPTS;
  float* pp = pk + img * KPTS;
  if (tid < KPTS) { b[tid] = 0.f; d[tid] = 0.f; pp[tid] = 0.f; }  // unfilled slots: p=0 => "absent"

  int c[10];
  #pragma unroll
  for (int L = 0; L < 10; ++L) c[L] = 0;
  unsigned char lp[64];
  for (int j = 0; j < 64; ++j) { int i = tid * 64 + j; unsigned char L = prs[i]; lp[j] = L; if (L > 0) c[L - 1]++; }
  #pragma unroll
  for (int L = 0; L < 10; ++L) cnt[L][tid] = (unsigned)c[L];
  __syncthreads();
  if (tid < 10) {
    unsigned run = 0;
    for (int t = 0; t < 256; ++t) { unsigned tmp = cnt[tid][t]; cnt[tid][t] = run; run += tmp; }
    total[tid] = run;
  }
  __syncthreads();
  if (tid == 0) {
    unsigned acc = 0;
    for (int L = 10; L >= 1; --L) { above[L] = acc; acc += total[L - 1]; }
  }
  __syncthreads();
  int c2[10];
  #pragma unroll
  for (int L = 0; L < 10; ++L) c2[L] = 0;
  for (int j = 0; j < 64; ++j) {
    int i = tid * 64 + j; int L = lp[j];
    if (L > 0) {
      unsigned pos = above[L] + cnt[L - 1][tid] + (unsigned)c2[L - 1]; c2[L - 1]++;
      if (pos < KPTS) {                 // slot order: desc persistence, asc index (== lax.top_k)
        b[pos]  = (float)lev[i] * 0.1f;
        d[pos]  = (float)dlv[i] * 0.1f;
        pp[pos] = (float)L * 0.1f;
      }
    }
  }
}

// -------- kernel 5: per-sample cost + 256-sum via V_WMMA_F32_16X16X4_F32 --------
__global__ void k5_wdist(const float* __restrict__ bk, const float* __restrict__ dk,
                         const float* __restrict__ pk, float* __restrict__ out) {
  __shared__ float cost[BATCH][KPTS];
  __shared__ float wsum[BATCH];
  int tid = threadIdx.x;
  int w = tid >> 5;          // one wave32 per sample
  int lane = tid & 31;
  const float* b1 = bk + w * KPTS;       const float* b2 = bk + (4 + w) * KPTS;
  const float* d1 = dk + w * KPTS;       const float* d2 = dk + (4 + w) * KPTS;
  const float* p1 = pk + w * KPTS;       const float* p2 = pk + (4 + w) * KPTS;
  for (int j = 0; j < 8; ++j) {
    int i = j * 32 + lane;
    float P1 = p1[i], P2 = p2[i];
    bool h1 = P1 > 0.f, h2 = P2 > 0.f;
    float cc = 0.f;
    if (h1 && h2)      { float db = b1[i] - b2[i], dd = d1[i] - d2[i]; cc = db * db + dd * dd; }
    else if (h1)       { cc = 0.5f * P1 * P1; }
    else if (h2)       { cc = 0.5f * P2 * P2; }
    cost[w][i] = cc;
  }
  __syncthreads();
  // Sum 256 f32 via matrix pipe: D = A(16x4) x ones(4x16) + C, 4 chunks of 64.
  // A layout (ISA 7.12.2, 32-bit A 16x4): v0 lanes0-15 K=0, lanes16-31 K=2; v1: K=1 / K=3.
  v8f acc = {0.f, 0.f, 0.f, 0.f, 0.f, 0.f, 0.f, 0.f};
  v2f ones; ones.x = 1.0f; ones.y = 1.0f;
  int off = (lane & 15) + ((lane >> 4) << 5);
  #pragma unroll
  for (int cch = 0; cch < 4; ++cch) {
    v2f a; a.x = cost[w][cch * 64 + off]; a.y = cost[w][cch * 64 + off + 16];
    acc = __builtin_amdgcn_wmma_f32_16x16x4_f32(false, a, false, ones, (short)0, acc,
                                                false, false);
  }
  // Column N of D: lane N holds M=0..7 (v0..v7), lane N+16 holds M=8..15.
  float part = acc[0] + acc[1] + acc[2] + acc[3] + acc[4] + acc[5] + acc[6] + acc[7];
  part += __shfl_xor(part, 16, 32);      // full 256-element sum in every lane
  if (lane == 0) wsum[w] = sqrtf(part + 1e-12f);
  __syncthreads();
  if (tid == 0) out[0] = 0.01f * (wsum[0] + wsum[1] + wsum[2] + wsum[3]) * 0.25f;
}

extern "C" void kernel_launch(void* const* d_in, const int* in_sizes, int n_in,
                              void* d_out, int out_size, void* d_ws, size_t ws_size,
                              hipStream_t stream) {
  (void)in_sizes; (void)n_in; (void)out_size; (void)ws_size;
  const float* mo = (const float*)d_in[0];   // model_output (4,1,128,128) f32
  const float* lb = (const float*)d_in[1];   // labels       (4,1,128,128) f32
  float* out = (float*)d_out;
  char* ws = (char*)d_ws;

  float*          mnmx   = (float*)(ws + WS_MNMX);
  int*            maxlev = (int*)(ws + WS_MAXLEV);
  unsigned char*  levg   = (unsigned char*)(ws + WS_LEV);
  unsigned char*  dlevg  = (unsigned char*)(ws + WS_DLEV);
  unsigned char*  persg  = (unsigned char*)(ws + WS_PERS);
  unsigned short* rankg  = (unsigned short*)(ws + WS_RANK);
  unsigned short* ordg   = (unsigned short*)(ws + WS_ORDER);
  float*          bkf    = (float*)(ws + WS_BK);
  float*          dkf    = (float*)(ws + WS_DK);
  float*          pkf    = (float*)(ws + WS_PK);

  k1_minmax   <<<BATCH, 256, 0, stream>>>(mo, mnmx);
  k2_quant_rank<<<NIMG, 256, 0, stream>>>(mo, lb, mnmx, levg, rankg, ordg, maxlev);
  k3_uf       <<<NIMG, 256, 4 * NPIX * sizeof(unsigned short), stream>>>(levg, rankg, ordg,
                                                                         maxlev, dlevg, persg);
  k4_topk     <<<NIMG, 256, 0, stream>>>(levg, dlevg, persg, bkf, dkf, pkf);
  k5_wdist    <<<1, 128, 0, stream>>>(bkf, dkf, pkf, out);
}